// GAT_15547781612146
// MI455X (gfx1250) — compile-verified
//
#include <hip/hip_runtime.h>
#include <hip/hip_bf16.h>

typedef float v2f __attribute__((ext_vector_type(2)));
typedef float v8f __attribute__((ext_vector_type(8)));

#define F_IN   128
#define F_HID  64
#define F_OUT  64
#define NHEAD  2
#define NEG_SLOPE 0.01f

// ---------------------------------------------------------------------------
// Zero-fill scratch (graph-capture safe, deterministic).
// ---------------------------------------------------------------------------
__global__ void gat_zero_kernel(float* __restrict__ p, size_t n) {
    size_t i = (size_t)blockIdx.x * blockDim.x + threadIdx.x;
    size_t stride = (size_t)gridDim.x * blockDim.x;
    for (; i < n; i += stride) p[i] = 0.0f;
}

// ---------------------------------------------------------------------------
// WMMA GEMM: C[:, coff + 0..63] = A[M,K] * B[K,64] + bias   (f32, exact)
// One 16x16 output tile per wave via V_WMMA_F32_16X16X4_F32, K stepped by 4.
// Block = 128 threads = 4 waves = the four 16-wide column tiles of the
// 64-wide output. gridDim.x = M/16 row tiles. EXEC is all-1s throughout the
// WMMA sequence (stores are predicated after the MMA loop).
//
// f32 16x4 A layout: lane l (l16 = l&15, half = l>>4): M = l16,
//   vgpr0 = K = k0 + 2*half, vgpr1 = K = k0 + 2*half + 1.
// B (4x16) mirrors with N = l16. C: vgpr r -> row r + 8*half, col l16.
// ---------------------------------------------------------------------------
__global__ __launch_bounds__(128) void gat_wmma_gemm_bias(
    const float* __restrict__ A, int lda,
    const float* __restrict__ B, int ldb,
    const float* __restrict__ bias,
    float* __restrict__ C, int ldc, int coff,
    int M, int K)
{
    const int wave = threadIdx.x >> 5;
    const int lane = threadIdx.x & 31;
    const int l16  = lane & 15;
    const int half = lane >> 4;

    const int m0 = blockIdx.x * 16;
    const int n0 = wave * 16;

    int arow = m0 + l16;
    if (arow >= M) arow = M - 1;               // clamp loads, keep EXEC full
    const float* __restrict__ Arow = A + (size_t)arow * lda;

    v8f c = {};
#pragma unroll 8
    for (int k0 = 0; k0 < K; k0 += 4) {
        const int ka = k0 + 2 * half;
        v2f a, b;
        a.x = Arow[ka];
        a.y = Arow[ka + 1];
        b.x = B[(size_t)ka * ldb + n0 + l16];
        b.y = B[(size_t)(ka + 1) * ldb + n0 + l16];
        c = __builtin_amdgcn_wmma_f32_16x16x4_f32(
                /*neg_a=*/false, a, /*neg_b=*/false, b,
                /*c_mod=*/(short)0, c, /*reuse_a=*/false, /*reuse_b=*/false);
    }

    const float bv = bias[n0 + l16];
#pragma unroll
    for (int r = 0; r < 8; ++r) {
        const int row = m0 + r + 8 * half;
        if (row < M)
            C[(size_t)row * ldc + coff + n0 + l16] = c[r] + bv;
    }
}

// ---------------------------------------------------------------------------
// Per-node score: s_src[n] = dot(H[n,:64], asrc), s_dst likewise.
// One wave32 per node, 2 floats per lane, shfl_xor tree reduction.
// ---------------------------------------------------------------------------
__global__ __launch_bounds__(256) void gat_score_kernel(
    const float* __restrict__ Hm, int ldh, int coff,
    const float* __restrict__ asrc, const float* __restrict__ adst,
    float* __restrict__ ssrc, float* __restrict__ sdst, int nnodes)
{
    const int wid  = (int)(((size_t)blockIdx.x * blockDim.x + threadIdx.x) >> 5);
    const int lane = threadIdx.x & 31;
    if (wid >= nnodes) return;

    const float* __restrict__ rowp = Hm + (size_t)wid * ldh + coff + 2 * lane;
    const float v0 = rowp[0], v1 = rowp[1];
    float ps = v0 * asrc[2 * lane] + v1 * asrc[2 * lane + 1];
    float pd = v0 * adst[2 * lane] + v1 * adst[2 * lane + 1];
#pragma unroll
    for (int off = 16; off >= 1; off >>= 1) {
        ps += __shfl_xor(ps, off);
        pd += __shfl_xor(pd, off);
    }
    if (lane == 0) { ssrc[wid] = ps; sdst[wid] = pd; }
}

// ---------------------------------------------------------------------------
// Per-edge score: e = leaky_relu(s_src[row] + s_dst[col]); e_all[col] += e.
// ---------------------------------------------------------------------------
__global__ __launch_bounds__(256) void gat_edge_score_kernel(
    const int* __restrict__ row, const int* __restrict__ col,
    const float* __restrict__ ssrc, const float* __restrict__ sdst,
    float* __restrict__ earr, float* __restrict__ eall, int nedges)
{
    const int e = blockIdx.x * blockDim.x + threadIdx.x;
    if (e >= nedges) return;
    float v = ssrc[row[e]] + sdst[col[e]];
    v = (v >= 0.0f) ? v : NEG_SLOPE * v;
    earr[e] = v;
    atomicAdd(&eall[col[e]], v);
}

// ---------------------------------------------------------------------------
// Per-edge scatter: out[col] += exp(e - e_all[col]) * H[row].
// One wave32 per edge: lanes cover the 64-float feature row (2 per lane),
// coalesced 256B gather + L2-resident float atomics.
// ---------------------------------------------------------------------------
__global__ __launch_bounds__(256) void gat_scatter_kernel(
    const int* __restrict__ row, const int* __restrict__ col,
    const float* __restrict__ earr, const float* __restrict__ eall,
    const float* __restrict__ Hm, int ldh, int hcoff,
    float* __restrict__ Out, int ldo, int ocoff, int nedges)
{
    const int wid  = (int)(((size_t)blockIdx.x * blockDim.x + threadIdx.x) >> 5);
    const int lane = threadIdx.x & 31;
    if (wid >= nedges) return;

    const int r = row[wid];
    const int c = col[wid];
    const float norm = expf(earr[wid] - eall[c]);

    const float* __restrict__ hr = Hm + (size_t)r * ldh + hcoff + 2 * lane;
    float* __restrict__ op = Out + (size_t)c * ldo + ocoff + 2 * lane;
    atomicAdd(op,     norm * hr[0]);
    atomicAdd(op + 1, norm * hr[1]);
}

// ---------------------------------------------------------------------------
// Final: out = log_softmax(elu(agg), axis=1). One wave32 per node.
// ---------------------------------------------------------------------------
__global__ __launch_bounds__(256) void gat_final_kernel(
    const float* __restrict__ agg, float* __restrict__ out, int nnodes)
{
    const int wid  = (int)(((size_t)blockIdx.x * blockDim.x + threadIdx.x) >> 5);
    const int lane = threadIdx.x & 31;
    if (wid >= nnodes) return;

    const float* __restrict__ rowp = agg + (size_t)wid * F_OUT + 2 * lane;
    float v0 = rowp[0], v1 = rowp[1];
    v0 = (v0 > 0.0f) ? v0 : (expf(v0) - 1.0f);
    v1 = (v1 > 0.0f) ? v1 : (expf(v1) - 1.0f);

    float m = fmaxf(v0, v1);
#pragma unroll
    for (int off = 16; off >= 1; off >>= 1) m = fmaxf(m, __shfl_xor(m, off));
    float s = expf(v0 - m) + expf(v1 - m);
#pragma unroll
    for (int off = 16; off >= 1; off >>= 1) s += __shfl_xor(s, off);
    const float ls = m + logf(s);

    float* __restrict__ op = out + (size_t)wid * F_OUT + 2 * lane;
    op[0] = v0 - ls;
    op[1] = v1 - ls;
}

// ---------------------------------------------------------------------------
extern "C" void kernel_launch(void* const* d_in, const int* in_sizes, int n_in,
                              void* d_out, int out_size, void* d_ws, size_t ws_size,
                              hipStream_t stream) {
    const float* x     = (const float*)d_in[0];
    const int*   ei    = (const int*)  d_in[1];
    const float* W1    = (const float*)d_in[2];   // [2,128,64]
    const float* b1    = (const float*)d_in[3];   // [2,64]
    const float* asrc1 = (const float*)d_in[4];   // [2,64]
    const float* adst1 = (const float*)d_in[5];   // [2,64]
    const float* W2    = (const float*)d_in[6];   // [128,64]
    const float* b2    = (const float*)d_in[7];   // [64]
    const float* asrc2 = (const float*)d_in[8];   // [64]
    const float* adst2 = (const float*)d_in[9];   // [64]

    const int N = in_sizes[0] / F_IN;             // 50000
    const int E = in_sizes[1] / 2;                // 800000
    const int* row = ei;
    const int* col = ei + E;

    // ---- workspace layout (floats) ----
    float* ws = (float*)d_ws;
    size_t o = 0;
    float* h1    = ws + o; o += (size_t)N * (NHEAD * F_HID);  // [N,128]
    // zero-filled region (contiguous):
    float* xcat  = ws + o; o += (size_t)N * (NHEAD * F_HID);  // agg1 -> layer2 input
    float* agg2  = ws + o; o += (size_t)N * F_OUT;
    float* eall1 = ws + o; o += (size_t)NHEAD * N;
    float* eall2 = ws + o; o += (size_t)N;
    const size_t zero_n = (size_t)N * (NHEAD * F_HID) + (size_t)N * F_OUT
                        + (size_t)(NHEAD + 1) * N;
    // non-zeroed scratch:
    float* h2    = ws + o; o += (size_t)N * F_OUT;
    float* ssrc1 = ws + o; o += (size_t)NHEAD * N;
    float* sdst1 = ws + o; o += (size_t)NHEAD * N;
    float* ssrc2 = ws + o; o += (size_t)N;
    float* sdst2 = ws + o; o += (size_t)N;
    float* e1    = ws + o; o += (size_t)NHEAD * E;
    float* e2    = ws + o; o += (size_t)E;
    (void)ws_size; (void)n_in; (void)out_size;

    gat_zero_kernel<<<2048, 256, 0, stream>>>(xcat, zero_n);

    const int tilesM     = (N + 15) / 16;
    const int nodeWaveBk = (N + 7) / 8;           // 8 waves / 256-thread block
    const int edgeWaveBk = (E + 7) / 8;
    const int edgeThrBk  = (E + 255) / 256;

    // ---- Layer 1 (two heads) ----
    for (int h = 0; h < NHEAD; ++h) {
        gat_wmma_gemm_bias<<<tilesM, 128, 0, stream>>>(
            x, F_IN, W1 + (size_t)h * F_IN * F_HID, F_HID, b1 + h * F_HID,
            h1, NHEAD * F_HID, h * F_HID, N, F_IN);
        gat_score_kernel<<<nodeWaveBk, 256, 0, stream>>>(
            h1, NHEAD * F_HID, h * F_HID,
            asrc1 + h * F_HID, adst1 + h * F_HID,
            ssrc1 + (size_t)h * N, sdst1 + (size_t)h * N, N);
        gat_edge_score_kernel<<<edgeThrBk, 256, 0, stream>>>(
            row, col, ssrc1 + (size_t)h * N, sdst1 + (size_t)h * N,
            e1 + (size_t)h * E, eall1 + (size_t)h * N, E);
        gat_scatter_kernel<<<edgeWaveBk, 256, 0, stream>>>(
            row, col, e1 + (size_t)h * E, eall1 + (size_t)h * N,
            h1, NHEAD * F_HID, h * F_HID,
            xcat, NHEAD * F_HID, h * F_HID, E);
    }

    // ---- Layer 2 ----
    gat_wmma_gemm_bias<<<tilesM, 128, 0, stream>>>(
        xcat, NHEAD * F_HID, W2, F_OUT, b2, h2, F_OUT, 0, N, NHEAD * F_HID);
    gat_score_kernel<<<nodeWaveBk, 256, 0, stream>>>(
        h2, F_OUT, 0, asrc2, adst2, ssrc2, sdst2, N);
    gat_edge_score_kernel<<<edgeThrBk, 256, 0, stream>>>(
        row, col, ssrc2, sdst2, e2, eall2, E);
    gat_scatter_kernel<<<edgeWaveBk, 256, 0, stream>>>(
        row, col, e2, eall2, h2, F_OUT, 0, agg2, F_OUT, 0, E);

    // ---- elu + log_softmax ----
    gat_final_kernel<<<nodeWaveBk, 256, 0, stream>>>(agg2, (float*)d_out, N);
}